// HamGNNInspiredNodeBlock_33818572489320
// MI455X (gfx1250) — compile-verified
//
#include <hip/hip_runtime.h>

typedef float v2f __attribute__((ext_vector_type(2)));
typedef float v8f __attribute__((ext_vector_type(8)));

namespace {

constexpr int XDIM = 208;   // total irrep dim per node
constexpr int XSTR = 209;   // padded LDS stride (odd -> conflict-free row access)
constexpr int WAVES = 4;    // waves per workgroup
constexpr int NPW   = 16;   // nodes per wave (WMMA M dimension)

// Factored CG representation:
//   s[p] = sum_i CG[t].v * xu[CG[t].i]   for terms t with pair index p
//   a[PR[p].k] += s[p] * xv[PR[p].j]
struct CGTerm { int i, p; float v; };
struct JKPair { int j, k; };

// ---- real-basis Wigner 3j values (unit Frobenius norm, e3nn order) ----
constexpr float R3 = 0.57735027f;   // 1/sqrt(3)
constexpr float R5 = 0.44721360f;   // 1/sqrt(5)
constexpr float CA = 0.31622776f;   // 1/sqrt(10)
constexpr float CB = 0.36514837f;   // 2/sqrt(30)
constexpr float CC = 0.18257418f;   // 1/sqrt(30)
constexpr float PA = 0.23904572f;
constexpr float PB = 0.11952286f;
constexpr float PC = 0.20701967f;

constexpr CGTerm CG000[1] = {{0,0,1.0f}};
constexpr JKPair PR000[1] = {{0,0}};

constexpr CGTerm CG011[3] = {{0,0,R3},{0,1,R3},{0,2,R3}};
constexpr JKPair PR011[3] = {{0,0},{1,1},{2,2}};

constexpr CGTerm CG101[3] = {{0,0,R3},{1,1,R3},{2,2,R3}};
constexpr JKPair PR101[3] = {{0,0},{0,1},{0,2}};

constexpr CGTerm CG110[3] = {{0,0,R3},{1,1,R3},{2,2,R3}};
constexpr JKPair PR110[3] = {{0,0},{1,0},{2,0}};

constexpr CGTerm CG022[5] = {{0,0,R5},{0,1,R5},{0,2,R5},{0,3,R5},{0,4,R5}};
constexpr JKPair PR022[5] = {{0,0},{1,1},{2,2},{3,3},{4,4}};

constexpr CGTerm CG202[5] = {{0,0,R5},{1,1,R5},{2,2,R5},{3,3,R5},{4,4,R5}};
constexpr JKPair PR202[5] = {{0,0},{0,1},{0,2},{0,3},{0,4}};

constexpr CGTerm CG220[5] = {{0,0,R5},{1,1,R5},{2,2,R5},{3,3,R5},{4,4,R5}};
constexpr JKPair PR220[5] = {{0,0},{1,0},{2,0},{3,0},{4,0}};

// 1 (x) 1 -> 2   (l=1 order: y,z,x ; l=2 order: xy,yz,z2,zx,x2-y2)
constexpr CGTerm CG112[11] = {
  {2,0,CA},{0,1,CA},{0,2,CA},{1,3,CA},{1,4,CB},{0,5,-CC},
  {2,6,-CC},{2,7,CA},{1,8,CA},{2,9,CA},{0,10,-CA}};
constexpr JKPair PR112[11] = {
  {0,0},{2,0},{1,1},{0,1},{1,2},{0,2},{2,2},{1,3},{2,3},{2,4},{0,4}};

// 1 (x) 2 -> 1  : W121[i,j,k] = W112[i,k,j]
constexpr CGTerm CG121[11] = {
  {2,0,CA},{0,1,CA},{0,2,CA},{1,3,CA},{1,4,CB},{0,5,-CC},
  {2,6,-CC},{2,7,CA},{1,8,CA},{2,9,CA},{0,10,-CA}};
constexpr JKPair PR121[11] = {
  {0,0},{0,2},{1,1},{1,0},{2,1},{2,0},{2,2},{3,1},{3,2},{4,2},{4,0}};

// 2 (x) 1 -> 1  : W211[i,j,k] = W112[j,k,i]  (pairs (0,0) and (2,2) merge 2 terms)
constexpr CGTerm CG211[11] = {
  {0,0,CA},{0,1,CA},{1,2,CA},{1,3,CA},{2,4,CB},
  {2,5,-CC},{4,5,-CA},{2,6,-CC},{4,6,CA},{3,7,CA},{3,8,CA}};
constexpr JKPair PR211[9] = {
  {2,0},{0,2},{0,1},{1,0},{1,1},{0,0},{2,2},{2,1},{1,2}};

// 2 (x) 2 -> 2  (25 terms, 23 distinct (j,k) pairs)
constexpr CGTerm CG222[25] = {
  {0,0,-PA},{0,1,-PA},{2,2,-PA},
  {1,3, PB},{1,4, PB},{2,5, PB},
  {3,6, PB},{3,7, PB},{2,8, PB},
  {4,9,-PA},{4,10,-PA},{2,11,-PA},
  {2,12, PA},
  {1,13,-PC},{1,14,-PC},{4,5,-PC},
  {3,15, PC},{3,16, PC},{4,8, PC},
  {0,17, PC},{0,18, PC},{1,19, PC},{1,20, PC},{3,21, PC},{3,22, PC}};
constexpr JKPair PR222[23] = {
  {0,2},{2,0},{0,0},{1,2},{2,1},{1,1},{3,2},{2,3},{3,3},{4,2},{2,4},{4,4},
  {2,2},{1,4},{4,1},{3,4},{4,3},{1,3},{3,1},{0,3},{3,0},{0,1},{1,0}};

// ---- one tensor-product path: out[z,w,k] += sum_uv T[z,uv,k] (coeff*W[uv,w]) ----
// u is chunk-uniform (M2 in {16,32}); per-u precompute s[p] = sum_i c*xu[i]
// amortized over M2/4 chunks; inner A element = 1 FMA per (j,k) pair.
// A tile (16 nodes x 4 uv) from LDS x-tile; B tile (4 uv x 16 w) from
// L2-resident weights; accumulate with V_WMMA_F32_16X16X4_F32.
template<int M1,int L1,int OFF1,int M2,int L2,int OFF2,int M3,int L3,
         int NCG, const CGTerm (&CG)[NCG], int NP, const JKPair (&PR)[NP]>
__device__ __forceinline__ void tp_path(const float* __restrict__ tpw, float coeff,
                                        const float* xrow, int lane,
                                        v8f (&acc)[M3/16][2*L3+1])
{
  constexpr int D1 = 2*L1+1, D2 = 2*L2+1, D3 = 2*L3+1, NT = M3/16;
  const int hi2  = (lane >> 4) << 1;   // lanes 16-31 hold K=2,3
  const int wcol = lane & 15;
  for (int u = 0; u < M1; ++u) {
    float xu[D1];
    #pragma unroll
    for (int i = 0; i < D1; ++i) xu[i] = xrow[OFF1 + u*D1 + i];
    float s[NP];
    #pragma unroll
    for (int p = 0; p < NP; ++p) s[p] = 0.0f;
    #pragma unroll
    for (int t = 0; t < NCG; ++t)
      s[CG[t].p] += (coeff * CG[t].v) * xu[CG[t].i];
    for (int v0 = 0; v0 < M2; v0 += 4) {
      float a[2][D3];
      #pragma unroll
      for (int q = 0; q < 2; ++q) {
        const int v = v0 + hi2 + q;
        float xv[D2];
        #pragma unroll
        for (int j = 0; j < D2; ++j) xv[j] = xrow[OFF2 + v*D2 + j];
        #pragma unroll
        for (int k = 0; k < D3; ++k) a[q][k] = 0.0f;
        #pragma unroll
        for (int p = 0; p < NP; ++p)
          a[q][PR[p].k] += s[p] * xv[PR[p].j];
      }
      const float* wp = tpw + (u*M2 + v0 + hi2) * M3 + wcol;
      #pragma unroll
      for (int nt = 0; nt < NT; ++nt) {
        v2f b;
        b.x = wp[nt*16];
        b.y = wp[nt*16 + M3];
        #pragma unroll
        for (int k = 0; k < D3; ++k) {
          v2f av; av.x = a[0][k]; av.y = a[1][k];
          acc[nt][k] = __builtin_amdgcn_wmma_f32_16x16x4_f32(
              false, av, false, b, (short)0, acc[nt][k], false, false);
        }
      }
    }
  }
}

// ---- o3.Linear skip: out[z,w,k] += (1/sqrt(M)) sum_u x[z,u,k] W[u,w] ----
template<int M,int L,int OFF>
__device__ __forceinline__ void lin_path(const float* __restrict__ lw, float scale,
                                         const float* xrow, int lane,
                                         v8f (&acc)[M/16][2*L+1])
{
  constexpr int D = 2*L+1, NT = M/16;
  const int hi2  = (lane >> 4) << 1;
  const int wcol = lane & 15;
  for (int u0 = 0; u0 < M; u0 += 4) {
    float a[2][D];
    #pragma unroll
    for (int q = 0; q < 2; ++q) {
      const int u = u0 + hi2 + q;
      #pragma unroll
      for (int k = 0; k < D; ++k) a[q][k] = scale * xrow[OFF + u*D + k];
    }
    const float* wp = lw + (u0 + hi2) * M + wcol;
    #pragma unroll
    for (int nt = 0; nt < NT; ++nt) {
      v2f b;
      b.x = wp[nt*16];
      b.y = wp[nt*16 + M];
      #pragma unroll
      for (int k = 0; k < D; ++k) {
        v2f av; av.x = a[0][k]; av.y = a[1][k];
        acc[nt][k] = __builtin_amdgcn_wmma_f32_16x16x4_f32(
            false, av, false, b, (short)0, acc[nt][k], false, false);
      }
    }
  }
}

// ---- store C tiles: VGPR r holds node r (lanes 0-15) / node r+8 (lanes 16-31) ----
template<int M,int L,int OFF>
__device__ __forceinline__ void store_irrep(float* __restrict__ out, long base, int lane,
                                            const v8f (&acc)[M/16][2*L+1])
{
  constexpr int D = 2*L+1, NT = M/16;
  const int mhi  = (lane >> 4) << 3;   // +8 rows for upper half-wave
  const int wcol = lane & 15;
  #pragma unroll
  for (int nt = 0; nt < NT; ++nt)
    #pragma unroll
    for (int k = 0; k < D; ++k)
      #pragma unroll
      for (int r = 0; r < 8; ++r) {
        const int node = r + mhi;
        const int col  = OFF + (nt*16 + wcol) * D + k;
        out[(base + node) * XDIM + col] = acc[nt][k][r];
      }
}

__global__ __launch_bounds__(WAVES * 32)
void e3nn_tp_wmma_kernel(const float* __restrict__ x,
                         const float* __restrict__ tpw,
                         const float* __restrict__ lw,
                         float* __restrict__ out,
                         int n_nodes)
{
  __shared__ float xs[WAVES][NPW * XSTR];
  const int wid  = threadIdx.x >> 5;
  const int lane = threadIdx.x & 31;
  const long base = ((long)blockIdx.x * WAVES + wid) * NPW;
  const bool active = (base + NPW) <= n_nodes;

  if (active) {
    // 16x208 node tile is contiguous in global memory -> coalesced loads
    const float* src = x + base * XDIM;
    for (int r = 0; r < NPW; ++r)
      for (int c = lane; c < XDIM; c += 32)
        xs[wid][r * XSTR + c] = src[r * XDIM + c];
  }
  __syncthreads();
  if (!active) return;

  const float* xrow = &xs[wid][(lane & 15) * XSTR];

  // path coefficients: sqrt((2l3+1)/fan_in)  (folded into hoisted s-precompute)
  constexpr float C0 = 0.020833333f;  // 1/48
  constexpr float C1 = 0.031250000f;  // 1/32
  constexpr float C2 = 0.046584750f;  // sqrt(5)/48

  // ---- phase 0: output irrep 32x0e (acc = 2 tiles, 16 VGPRs) ----
  {
    v8f acc0[2][1];
    #pragma unroll
    for (int t = 0; t < 2; ++t) acc0[t][0] = {};
    lin_path<32,0,0>(lw + 0, 0.17677670f, xrow, lane, acc0);      // 1/sqrt(32)
    tp_path<32,0,0,  32,0,0,  32,0, 1,CG000,1,PR000>(tpw + 0,      C0, xrow, lane, acc0);
    tp_path<32,1,32, 32,1,32, 32,0, 3,CG110,3,PR110>(tpw + 106496, C0, xrow, lane, acc0);
    tp_path<16,2,128,16,2,128,32,0, 5,CG220,5,PR220>(tpw + 196608, C0, xrow, lane, acc0);
    store_irrep<32,0,0>(out, base, lane, acc0);
  }

  // ---- phase 1: output irrep 32x1o (acc = 6 tiles, 48 VGPRs) ----
  {
    v8f acc1[2][3];
    #pragma unroll
    for (int t = 0; t < 2; ++t)
      #pragma unroll
      for (int k = 0; k < 3; ++k) acc1[t][k] = {};
    lin_path<32,1,32>(lw + 1024, 0.17677670f, xrow, lane, acc1);
    tp_path<32,0,0,  32,1,32, 32,1, 3,CG011, 3,PR011>(tpw + 32768,  C1, xrow, lane, acc1);
    tp_path<32,1,32, 32,0,0,  32,1, 3,CG101, 3,PR101>(tpw + 73728,  C1, xrow, lane, acc1);
    tp_path<32,1,32, 16,2,128,32,1,11,CG121,11,PR121>(tpw + 155648, C1, xrow, lane, acc1);
    tp_path<16,2,128,32,1,32, 32,1,11,CG211, 9,PR211>(tpw + 180224, C1, xrow, lane, acc1);
    store_irrep<32,1,32>(out, base, lane, acc1);
  }

  // ---- phase 2: output irrep 16x2e (acc = 5 tiles, 40 VGPRs) ----
  {
    v8f acc2[1][5];
    #pragma unroll
    for (int k = 0; k < 5; ++k) acc2[0][k] = {};
    lin_path<16,2,128>(lw + 2048, 0.25f, xrow, lane, acc2);       // 1/sqrt(16)
    tp_path<32,0,0,  16,2,128,16,2, 5,CG022, 5,PR022>(tpw + 65536,  C2, xrow, lane, acc2);
    tp_path<32,1,32, 32,1,32, 16,2,11,CG112,11,PR112>(tpw + 139264, C2, xrow, lane, acc2);
    tp_path<16,2,128,32,0,0,  16,2, 5,CG202, 5,PR202>(tpw + 172032, C2, xrow, lane, acc2);
    tp_path<16,2,128,16,2,128,16,2,25,CG222,23,PR222>(tpw + 204800, C2, xrow, lane, acc2);
    store_irrep<16,2,128>(out, base, lane, acc2);
  }
}

} // namespace

extern "C" void kernel_launch(void* const* d_in, const int* in_sizes, int n_in,
                              void* d_out, int out_size, void* d_ws, size_t ws_size,
                              hipStream_t stream) {
  (void)n_in; (void)out_size; (void)d_ws; (void)ws_size;
  const float* x   = (const float*)d_in[0];
  const float* tpw = (const float*)d_in[1];
  const float* lw  = (const float*)d_in[2];
  float* out = (float*)d_out;

  const int n_nodes = in_sizes[0] / XDIM;           // 200000
  const int tiles   = (n_nodes + NPW - 1) / NPW;    // 12500
  const int blocks  = (tiles + WAVES - 1) / WAVES;  // 3125

  hipLaunchKernelGGL(e3nn_tp_wmma_kernel, dim3(blocks), dim3(WAVES * 32), 0, stream,
                     x, tpw, lw, out, n_nodes);
}